// HilbertHuangTransform_73366631350909
// MI455X (gfx1250) — compile-verified
//
#include <hip/hip_runtime.h>

#define T 4096
#define NTHREADS 256
#define EMD_NUM_IMFS 8
#define EMD_SIFT_ITERS 10
#define EMD_SD_THRESH 0.2f
#define EMD_EPS 1e-12f

typedef float v2f __attribute__((ext_vector_type(2)));
typedef float v8f __attribute__((ext_vector_type(8)));

#define PI_F 3.14159265358979323846f
#define TWO_PI_F 6.28318530717958647692f

// base-16 digit reversal of a 3-digit (12-bit) index
__device__ __forceinline__ int rev12(int r) {
  return ((r & 15) << 8) | (r & 240) | (r >> 8);
}

__device__ __forceinline__ v8f wmma_f32(v2f a, v2f b, v8f c) {
  // D(16x16,f32) = A(16x4,f32) x B(4x16,f32) + C  -> v_wmma_f32_16x16x4_f32
  return __builtin_amdgcn_wmma_f32_16x16x4_f32(false, a, false, b, (short)0, c,
                                               false, false);
}

// ---------------------------------------------------------------------------
// In-place 4096-point complex FFT (forward, unnormalized), three radix-16
// stages, each 16-point DFT realized as W16(16x16) x X(16x256) with
// V_WMMA_F32_16X16X4_F32.  Data n = n0 + 16*n1 + 256*n2; result at flat
// r = 256*k0 + 16*k1 + k2  holds X[k0 + 16*k1 + 256*k2]  (digit-reversed).
// twc/tws hold W16[m][p] = cis(-2*pi*m*p/16).
// Per stage, each wave owns tiles {wave, wave+8}; a tile's input and output
// address sets coincide, so in-place is safe; barrier between stages.
// ---------------------------------------------------------------------------
__device__ __forceinline__ void fft4096(float* fr, float* fi,
                                        const float* twc, const float* tws,
                                        int lane, int wave) {
  const int n = lane & 15;          // column within tile (WMMA N) / A-row (M)
  const int half = (lane >> 4) & 1; // lane half selects K offset +2
  #pragma unroll
  for (int st = 0; st < 3; ++st) {
    for (int t_idx = wave; t_idx < 16; t_idx += 8) {
      int base, stride;
      float thmul; // twiddle angle = thmul * p (applied as multiply by e^{-i*theta})
      if (st == 0) {        // transform over n2 (stride 256), no twiddle
        base = t_idx * 16 + n; stride = 256; thmul = 0.0f;
      } else if (st == 1) { // transform over n1; k0 = t_idx; tw = W256^{p*k0}
        base = n + 256 * t_idx; stride = 16;
        thmul = (TWO_PI_F / 256.0f) * (float)t_idx;
      } else {              // transform over n0; tw = W4096^{p*(k0+16*k1)}
        base = 16 * n + 256 * t_idx; stride = 1;
        thmul = (TWO_PI_F / 4096.0f) * (float)(t_idx + 16 * n);
      }
      v8f accRe = {0.f, 0.f, 0.f, 0.f, 0.f, 0.f, 0.f, 0.f};
      v8f accIm = {0.f, 0.f, 0.f, 0.f, 0.f, 0.f, 0.f, 0.f};
      #pragma unroll
      for (int kk = 0; kk < 4; ++kk) {
        const int pA = 4 * kk + 2 * half; // K row for .x
        const int pB = pA + 1;            // K row for .y
        float xr = fr[base + stride * pA];
        float xi = fi[base + stride * pA];
        float yr = fr[base + stride * pB];
        float yi = fi[base + stride * pB];
        float sA, cA, sB, cB;
        __sincosf(thmul * (float)pA, &sA, &cA);
        __sincosf(thmul * (float)pB, &sB, &cB);
        v2f Bre, Bim;
        Bre.x = xr * cA + xi * sA;  Bim.x = xi * cA - xr * sA; // * e^{-i th}
        Bre.y = yr * cB + yi * sB;  Bim.y = yi * cB - yr * sB;
        v2f Are, Aim, AimN;
        Are.x = twc[n * 16 + pA];   Are.y = twc[n * 16 + pB];
        Aim.x = tws[n * 16 + pA];   Aim.y = tws[n * 16 + pB];
        AimN.x = -Aim.x;            AimN.y = -Aim.y;
        accRe = wmma_f32(Are,  Bre, accRe);  // + Wre*Xre
        accRe = wmma_f32(AimN, Bim, accRe);  // - Wim*Xim
        accIm = wmma_f32(Are,  Bim, accIm);  // + Wre*Xim
        accIm = wmma_f32(Aim,  Bre, accIm);  // + Wim*Xre
      }
      #pragma unroll
      for (int g = 0; g < 8; ++g) {  // D: vgpr g -> row M=g (+8 for lanes>=16)
        int q = g + 8 * half;
        int idx = base + stride * q;
        fr[idx] = accRe[g];
        fi[idx] = accIm[g];
      }
    }
    __syncthreads();
  }
}

// Count (and optionally compact in ascending order) strict local extrema of h.
__device__ __forceinline__ int scan_extrema(const float* h, int* kx, float* kv,
                                            int* cnt, int tid, int isMax,
                                            int doWrite) {
  int c = 0;
  const int base = tid * (T / NTHREADS);
  #pragma unroll
  for (int j = 0; j < T / NTHREADS; ++j) {
    int i = base + j;
    if (i >= 1 && i <= T - 2) {
      float v = h[i];
      bool e = isMax ? (v > h[i - 1] && v > h[i + 1])
                     : (v < h[i - 1] && v < h[i + 1]);
      c += e ? 1 : 0;
    }
  }
  cnt[tid] = c;
  __syncthreads();
  for (int off = 1; off < NTHREADS; off <<= 1) { // Hillis-Steele inclusive scan
    int v = (tid >= off) ? cnt[tid - off] : 0;
    __syncthreads();
    cnt[tid] += v;
    __syncthreads();
  }
  int total = cnt[NTHREADS - 1];
  if (doWrite) {
    int wr = cnt[tid] - c; // exclusive offset
    for (int j = 0; j < T / NTHREADS; ++j) {
      int i = base + j;
      if (i >= 1 && i <= T - 2) {
        float v = h[i];
        bool e = isMax ? (v > h[i - 1] && v > h[i + 1])
                       : (v < h[i - 1] && v < h[i + 1]);
        if (e) { kx[wr] = i; kv[wr] = v; ++wr; }
      }
    }
  }
  __syncthreads();
  return total;
}

// env[i] (+)= scale * linear-interp through knots (edge-clamped), matching the
// reference's masked-knot jnp.interp semantics.
__device__ __forceinline__ void add_envelope(const float* h, const int* kx,
                                             const float* kv, int nk,
                                             float* env, float scale, int tid,
                                             bool first) {
  for (int i = tid; i < T; i += NTHREADS) {
    float v;
    if (nk == 0)      v = h[0];
    else if (nk == 1) v = kv[0];
    else if (i <= kx[0])      v = kv[0];
    else if (i >= kx[nk - 1]) v = kv[nk - 1];
    else {
      int lo = 0, hi = nk - 1;   // invariant: kx[lo] <= i < kx[hi]
      while (hi - lo > 1) {
        int mid = (lo + hi) >> 1;
        if (kx[mid] <= i) lo = mid; else hi = mid;
      }
      float f = (float)(i - kx[lo]) / (float)(kx[hi] - kx[lo]);
      v = kv[lo] + (kv[hi] - kv[lo]) * f;
    }
    if (first) env[i] = scale * v; else env[i] += scale * v;
  }
}

__device__ __forceinline__ float block_sum(float v, float* red, int tid) {
  red[tid] = v;
  __syncthreads();
  for (int off = NTHREADS >> 1; off > 0; off >>= 1) {
    if (tid < off) red[tid] += red[tid + off];
    __syncthreads();
  }
  float r = red[0];
  __syncthreads();
  return r;
}

// ---------------------------------------------------------------------------
// One workgroup per signal: EMD (8 IMFs x 10 sifts) + per-IMF Hilbert
// transform, fully LDS-resident; output streamed to HBM exactly once.
// ---------------------------------------------------------------------------
__global__ __launch_bounds__(NTHREADS) void
HilbertHuangTransform_73366631350909_kernel(const float* __restrict__ x,
                                            float* __restrict__ out) {
  __shared__ float s_res[T];      // running residue
  __shared__ float s_h[T];        // sift buffer / phase scratch
  __shared__ float s_env[T];      // envelope mean / permute scratch
  __shared__ float s_fr[T];       // FFT real
  __shared__ float s_fi[T];       // FFT imag
  __shared__ int   s_kx[T / 2];   // knot positions
  __shared__ float s_kv[T / 2];   // knot values
  __shared__ int   s_cnt[NTHREADS];
  __shared__ float s_red[NTHREADS];
  __shared__ float s_twc[256];    // W16 cos table
  __shared__ float s_tws[256];    // W16 sin table (of -2*pi*m*p/16)
  __shared__ int   s_flag[1];

  const int tid = threadIdx.x;
  const int sid = blockIdx.x;
  const int lane = tid & 31;
  const int wave = tid >> 5;

  { // W16[m][p] = cis(-2*pi*m*p/16)
    int mm = tid >> 4, pp = tid & 15;
    float s, c;
    __sincosf(-TWO_PI_F * (float)(mm * pp) / 16.0f, &s, &c);
    s_twc[tid] = c;
    s_tws[tid] = s;
  }
  for (int i = tid; i < T; i += NTHREADS) s_res[i] = x[(size_t)sid * T + i];
  __syncthreads();

  int outer_alive = 1; // block-uniform
  for (int m = 0; m < EMD_NUM_IMFS; ++m) {
    for (int i = tid; i < T; i += NTHREADS) s_h[i] = s_res[i];
    __syncthreads();

    // ---- sifting (dead iterations are exact no-ops in the reference) ----
    int alive = outer_alive;
    for (int it = 0; it < EMD_SIFT_ITERS && alive; ++it) {
      int nmax = scan_extrema(s_h, s_kx, s_kv, s_cnt, tid, 1, 0);
      int nmin = scan_extrema(s_h, s_kx, s_kv, s_cnt, tid, 0, 0);
      if (nmax < 2 || nmin < 2) { alive = 0; break; }
      int nk = scan_extrema(s_h, s_kx, s_kv, s_cnt, tid, 1, 1);
      add_envelope(s_h, s_kx, s_kv, nk, s_env, 0.5f, tid, true);
      __syncthreads();
      nk = scan_extrema(s_h, s_kx, s_kv, s_cnt, tid, 0, 1);
      add_envelope(s_h, s_kx, s_kv, nk, s_env, 0.5f, tid, false);
      __syncthreads();
      float pn = 0.f, pd = 0.f;
      for (int i = tid; i < T; i += NTHREADS) {
        float mi = s_env[i];   // h - h_new
        float hi = s_h[i];
        pn += mi * mi;
        pd += hi * hi;
        s_h[i] = hi - mi;      // act is true here -> always update
      }
      float num = block_sum(pn, s_red, tid);
      float den = block_sum(pd, s_red, tid);
      alive = ((num / (den + EMD_EPS)) >= EMD_SD_THRESH) ? 1 : 0;
    }
    __syncthreads();

    // ---- IMF out, residue update, seed FFT input ----
    const int cur_alive = outer_alive;
    for (int i = tid; i < T; i += NTHREADS) {
      float imf = cur_alive ? s_h[i] : 0.0f;
      out[((size_t)sid * 16 + m) * T + i] = imf;
      s_res[i] -= imf;
      s_fr[i] = imf;
      s_fi[i] = 0.0f;
    }
    __syncthreads();

    // ---- extrema count of new residue -> outer_alive for next IMF ----
    {
      int c = 0;
      const int base = tid * (T / NTHREADS);
      for (int j = 0; j < T / NTHREADS; ++j) {
        int i = base + j;
        if (i <= T - 3) {
          float d0 = s_res[i + 1] - s_res[i];
          float d1 = s_res[i + 2] - s_res[i + 1];
          int sg0 = (d0 > 0.f) - (d0 < 0.f);
          int sg1 = (d1 > 0.f) - (d1 < 0.f);
          c += (sg0 != sg1) ? 1 : 0;
        }
      }
      s_cnt[tid] = c;
      __syncthreads();
      if (tid == 0) {
        int tot = 0;
        for (int k = 0; k < NTHREADS; ++k) tot += s_cnt[k];
        s_flag[0] = tot;
      }
      __syncthreads();
      outer_alive = (s_flag[0] >= 2) ? outer_alive : 0;
      __syncthreads();
    }

    // ---- Hilbert: analytic signal via FFT -> mask -> conj-trick inverse ----
    float* fo = out + ((size_t)sid * 16 + 8 + m) * T;
    if (cur_alive) {
      fft4096(s_fr, s_fi, s_twc, s_tws, lane, wave);
      // Build conj(mask(k)*X[k]) in natural order (X[k] lives at rev(k)).
      for (int k = tid; k < T; k += NTHREADS) {
        int r = rev12(k);
        float mk = (k == 0 || k == T / 2) ? 1.0f : (k < T / 2 ? 2.0f : 0.0f);
        s_h[k] = mk * s_fr[r];
        s_env[k] = -mk * s_fi[r];
      }
      __syncthreads();
      for (int i = tid; i < T; i += NTHREADS) { s_fr[i] = s_h[i]; s_fi[i] = s_env[i]; }
      __syncthreads();
      fft4096(s_fr, s_fi, s_twc, s_tws, lane, wave);
      // analytic[n] = conj(result at rev(n))/T ; scale cancels in atan2
      for (int nn = tid; nn < T; nn += NTHREADS) {
        int r = rev12(nn);
        s_h[nn] = atan2f(-s_fi[r], s_fr[r]);
      }
      __syncthreads();
      // diff(unwrap(phase)) == wrap-to-(-pi,pi] of raw phase difference
      for (int nn = tid; nn < T; nn += NTHREADS) {
        int a = (nn < T - 1) ? nn : T - 2;
        float d = s_h[a + 1] - s_h[a];
        float w = d - TWO_PI_F * floorf((d + PI_F) / TWO_PI_F);
        fo[nn] = w / TWO_PI_F;
      }
      __syncthreads();
    } else {
      for (int nn = tid; nn < T; nn += NTHREADS) fo[nn] = 0.0f;
      __syncthreads();
    }
  }
}

extern "C" void kernel_launch(void* const* d_in, const int* in_sizes, int n_in,
                              void* d_out, int out_size, void* d_ws,
                              size_t ws_size, hipStream_t stream) {
  (void)n_in; (void)out_size; (void)d_ws; (void)ws_size;
  const float* x = (const float*)d_in[0];
  float* out = (float*)d_out;
  const int nsig = in_sizes[0] / T; // (32*64*4096)/4096 = 2048 signals
  HilbertHuangTransform_73366631350909_kernel<<<nsig, NTHREADS, 0, stream>>>(x, out);
}